// AtlasDetection_45646912422590
// MI455X (gfx1250) — compile-verified
//
#include <hip/hip_runtime.h>

typedef float v2f __attribute__((ext_vector_type(2)));
typedef float v8f __attribute__((ext_vector_type(8)));

#define NXv 160
#define NYv 160
#define NZv 64
#define NVOX (NXv * NYv * NZv)   // 1,638,400
#define BATCH 2
#define CH 32
#define FH 120
#define FW 160
#define VOXEL_SIZE 0.04f
#define INV_STRIDE 0.25f

// One lane per (batch, voxel). Each wave handles 32 consecutive voxels of one
// batch. The 3x4 projection matvec runs as V_WMMA_F32_16X16X4_F32:
//   A = scaled projection padded to 16x4, B = homogeneous world coords (4x16),
//   two WMMAs cover the wave's 32 voxels. Camera x/y/z = D rows 0/1/2.
//
// Key structural facts exploited:
//  * vbase is a multiple of 32 and NZ=64, so (v >> 6) — and therefore gx, gy,
//    wx, wy — is identical for ALL lanes of a wave: readfirstlane pushes the
//    div/mod-by-160 onto the SALU.
//  * gz never wraps inside a wave: gz = (vbase & 63) + j, j in [0,31].
//  * k0 is uniform per half-wave, so the B-matrix build is 2 cndmasks/tile.
__global__ __launch_bounds__(256)
void atlas_backproject(const float* __restrict__ proj,
                       const float* __restrict__ feat,
                       const float* __restrict__ origin,
                       float* __restrict__ vol,
                       float* __restrict__ validOut)
{
    const int lane = threadIdx.x & 31;
    const unsigned wave  = (blockIdx.x * blockDim.x + threadIdx.x) >> 5;
    const unsigned start = wave * 32u;              // linear (batch,voxel) start

    // Wave-uniform scalars -> force SALU with readfirstlane.
    const int      b     = (int)__builtin_amdgcn_readfirstlane(start / NVOX);
    const unsigned vbase = __builtin_amdgcn_readfirstlane(start % NVOX);
    const unsigned t     = vbase >> 6;              // uniform: same for all 32 voxels
    const int      gy    = (int)(t % NYv);          // SALU magic-mul
    const int      gx    = (int)(t / NYv);
    const int      zb    = (int)(vbase & 63u);      // z base; zb + 31 <= 63 (no wrap)

    const float ox = origin[0], oy = origin[1], oz = origin[2];
    const float wx = gx * VOXEL_SIZE + ox;          // wave-uniform world x/y
    const float wy = gy * VOXEL_SIZE + oy;

    // ---- A matrix (16x4 f32): lanes 0-15 hold {K=0,K=1}, lanes 16-31 {K=2,K=3} ----
    const int  m  = lane & 15;
    const bool lo = (lane < 16);
    const int  k0 = lo ? 0 : 2;
    v2f A = {0.0f, 0.0f};
    if (m < 3) {
        const float rs = (m == 2) ? 1.0f : INV_STRIDE;   // row scale = [1/S, 1/S, 1]
        const float* pr = proj + b * 12 + m * 4;
        A.x = pr[k0]     * rs;
        A.y = pr[k0 + 1] * rs;
    }

    // ---- B matrices (4x16 f32): column = lane&15; rows {0,1}=wx,wy on lanes 0-15,
    //      rows {2,3}=wz,1 on lanes 16-31.  wz only consumed by lanes >= 16. ----
    const int   col = lane & 15;
    const float wz0 = (zb + col)      * VOXEL_SIZE + oz;   // tile 0: z = zb+0..15
    const float wz1 = (zb + col + 16) * VOXEL_SIZE + oz;   // tile 1: z = zb+16..31
    v2f B0, B1;
    B0.x = lo ? wx : wz0;  B0.y = lo ? wy : 1.0f;
    B1.x = lo ? wx : wz1;  B1.y = lo ? wy : 1.0f;

    v8f c0 = {};
    v8f c1 = {};
    // (neg_a, A, neg_b, B, c_mod, C, reuse_a, reuse_b)
    c0 = __builtin_amdgcn_wmma_f32_16x16x4_f32(false, A, false, B0, (short)0, c0, false, false);
    c1 = __builtin_amdgcn_wmma_f32_16x16x4_f32(false, A, false, B1, (short)0, c1, false, false);

    // D layout: lanes 0-15, VGPR r = row r, column = lane. Rows 0/1/2 = cam x/y/z.
    // Tile-0 results stay on lanes 0-15; tile-1 columns move to lanes 16-31.
    float cx1 = __shfl(c1[0], col, 32);
    float cy1 = __shfl(c1[1], col, 32);
    float cz1 = __shfl(c1[2], col, 32);
    float cx = lo ? c0[0] : cx1;
    float cy = lo ? c0[1] : cy1;
    float cz = lo ? c0[2] : cz1;

    const unsigned v = vbase + (unsigned)lane;      // this lane's voxel

    // Perspective divide (exact IEEE, matches jnp round-to-even semantics).
    int px = (int)rintf(cx / cz);
    int py = (int)rintf(cy / cz);
    const bool ok = (px >= 0) & (py >= 0) & (px < FW) & (py < FH) & (cz > 0.0f);
    const int cpx = min(max(px, 0), FW - 1);
    const int cpy = min(max(py, 0), FH - 1);
    const int idx = cpy * FW + cpx;
    const float vf = ok ? 1.0f : 0.0f;

    validOut[(size_t)b * NVOX + v] = vf;

    // Gather 32 channels (feature maps are L2-resident: ~4.9 MB total),
    // stores fully coalesced along the voxel (z-fastest) axis.
    const float* fb = feat + (size_t)b * CH * (FH * FW) + idx;
    float*       ob = vol  + (size_t)b * CH * (size_t)NVOX + v;
#pragma unroll
    for (int c = 0; c < CH; ++c) {
        ob[(size_t)c * NVOX] = fb[c * (FH * FW)] * vf;
    }
}

extern "C" void kernel_launch(void* const* d_in, const int* in_sizes, int n_in,
                              void* d_out, int out_size, void* d_ws, size_t ws_size,
                              hipStream_t stream) {
    const float* proj   = (const float*)d_in[0];   // (B,3,4)
    const float* feat   = (const float*)d_in[1];   // (B,C,H,W)
    const float* origin = (const float*)d_in[2];   // (1,3)

    float* vol      = (float*)d_out;                           // (B,C,NX,NY,NZ)
    float* validOut = vol + (size_t)BATCH * CH * (size_t)NVOX; // (B,1,NX,NY,NZ)

    const int total  = BATCH * NVOX;   // 3,276,800 threads, one voxel each
    const int block  = 256;            // 8 full wave32s -> EXEC all-1s for WMMA
    const int blocks = total / block;  // 12,800 (exact)

    atlas_backproject<<<blocks, block, 0, stream>>>(proj, feat, origin, vol, validOut);
}